// LargeEmbeddingLookup_72292889526909
// MI455X (gfx1250) — compile-verified
//
#include <hip/hip_runtime.h>
#include <stdint.h>

// ---------------- problem constants (from the reference) ----------------
#define B_TOTAL 16384
#define LMAX    50
#define V0N     1000000
#define D0N     64
#define V1N     100000
#define D1N     128
#define OUTD    (D0N + D1N)   // 192

#define WAVES_PER_BLOCK 8
#define CHUNK 8                    // rows per TDM gather descriptor (32-bit index mode max)
#define BUF_ELEMS (CHUNK * 128)    // LDS tile sized for the widest table (D=128): 4KB

#if defined(__HIP_DEVICE_COMPILE__) && __has_builtin(__builtin_amdgcn_tensor_load_to_lds) && __has_builtin(__builtin_amdgcn_s_wait_tensorcnt)
#define USE_TDM 1
#else
#define USE_TDM 0
#endif

typedef __attribute__((ext_vector_type(4))) unsigned int v4u;
typedef __attribute__((ext_vector_type(8))) int v8i;
typedef __attribute__((ext_vector_type(4))) int v4i;

__device__ __forceinline__ int rfl(int x) {
#if defined(__HIP_DEVICE_COMPILE__)
  return __builtin_amdgcn_readfirstlane(x);
#else
  return x;
#endif
}

#if USE_TDM
// Issue one TDM gather-mode descriptor: fetch cnt = min(8, len-l0) rows
// (row i = idxrow[l0+i]) of a [V x D] f32 tensor at E into LDS at byte offset
// lds_off, rows compacted contiguously. Caller guarantees len >= 1.
template<int D>
__device__ __forceinline__ void tdm_issue_gather(const float* __restrict__ E, int V,
                                                 const int* __restrict__ idxrow,
                                                 int l0, int len, unsigned lds_off)
{
  int id[CHUNK];
#pragma unroll
  for (int j = 0; j < CHUNK; ++j) {
    int l = l0 + j;
    int lc = (l < len) ? l : (len - 1);   // branch-free clamp; dup tails masked by tile_dim1
    id[j] = rfl(idxrow[lc]);
  }
  int cnt = len - l0; if (cnt > CHUNK) cnt = CHUNK;
  cnt = rfl(cnt);

  uint64_t ga = (uint64_t)(uintptr_t)E;
  // ---- D# group 0 (128b): mode bits | lds_addr | global_addr | type ----
  v4u g0;
  g0.x = 0xC0000001u;                                      // count=1, gather_index_size=32b, gather_mode=1
  g0.y = lds_off;                                          // LDS byte offset of tile start
  g0.z = (unsigned)(ga & 0xFFFFFFFFu);                     // global_addr[31:0]
  g0.w = (unsigned)((ga >> 32) & 0x01FFFFFFu) | (2u << 30);// global_addr[56:32] | type=2
  // ---- D# group 1 (256b): data_size/dims/strides ----
  v8i g1;
  g1[0] = (int)(2u << 16);                                            // data_size = 4 bytes
  g1[1] = (int)(((unsigned)D & 0xFFFFu) << 16);                       // tensor_dim0[15:0]
  g1[2] = (int)((((unsigned)V & 0xFFFFu) << 16) | ((unsigned)D >> 16)); // tensor_dim1 lo16 | tensor_dim0 hi16
  g1[3] = (int)(((unsigned)D << 16) | ((unsigned)V >> 16));           // tile_dim0 | tensor_dim1 hi16
  g1[4] = cnt & 0xFFFF;                                               // tile_dim1 = #valid gather indices
  g1[5] = D;                                                          // tensor_dim0_stride (elements)
  g1[6] = 0;
  g1[7] = 0;
  // ---- D# groups 2+3: the 8 row indices (32-bit index mode) ----
  v4i g2 = { id[0], id[1], id[2], id[3] };
  v4i g3 = { id[4], id[5], id[6], id[7] };

#if defined(__clang_major__) && (__clang_major__ >= 23)
  v8i g4 = { 0, 0, 0, 0, 0, 0, 0, 0 };
  __builtin_amdgcn_tensor_load_to_lds(g0, g1, g2, g3, g4, 0);
#else
  __builtin_amdgcn_tensor_load_to_lds(g0, g1, g2, g3, 0);
#endif
}
#endif // USE_TDM

__global__ __launch_bounds__(WAVES_PER_BLOCK * 32)
void embbag_tdm_kernel(const int* __restrict__ idx0, const int* __restrict__ len0,
                       const int* __restrict__ idx1, const int* __restrict__ len1,
                       const float* __restrict__ E0, const float* __restrict__ E1,
                       float* __restrict__ out)
{
  __shared__ __align__(16) float smem[WAVES_PER_BLOCK][2][BUF_ELEMS];  // 64 KB/block

  const int wave = threadIdx.x >> 5;
  const int lane = threadIdx.x & 31;
  const int b = blockIdx.x * WAVES_PER_BLOCK + wave;
  if (b >= B_TOTAL) return;

  float* buf0 = &smem[wave][0][0];
  float* buf1 = &smem[wave][1][0];
  const unsigned off0 = (unsigned)(uintptr_t)buf0;  // flat LDS addr low 32b == LDS byte offset
  const unsigned off1 = (unsigned)(uintptr_t)buf1;
  (void)off0; (void)off1;

  const int* idxr0 = idx0 + (size_t)b * LMAX;
  const int* idxr1 = idx1 + (size_t)b * LMAX;
  int l0n = rfl(len0[b]); if (l0n < 0) l0n = 0; if (l0n > LMAX) l0n = LMAX;
  int l1n = rfl(len1[b]); if (l1n < 0) l1n = 0; if (l1n > LMAX) l1n = LMAX;

  // table-0 accumulator: half-wave pair covers 2 rows/pass over 64 columns
  const int s0 = lane & 15;    // float4 column within a D=64 row
  const int r0 = lane >> 4;    // row parity within a pass
  float a0x = 0.f, a0y = 0.f, a0z = 0.f, a0w = 0.f;
  // table-1 accumulator: whole wave covers one D=128 row per pass
  float a1x = 0.f, a1y = 0.f, a1z = 0.f, a1w = 0.f;

#if USE_TDM
  const int nch0 = (l0n + CHUNK - 1) / CHUNK;
  const int nch1 = (l1n + CHUNK - 1) / CHUNK;
  const int total = nch0 + nch1;

  if (total > 0) {
    // Unified chunk stream across both tables: no TDM pipeline drain between tables.
    auto issue = [&](int t) {
      const unsigned off = (t & 1) ? off1 : off0;
      if (t < nch0) tdm_issue_gather<D0N>(E0, V0N, idxr0, t * CHUNK, l0n, off);
      else          tdm_issue_gather<D1N>(E1, V1N, idxr1, (t - nch0) * CHUNK, l1n, off);
    };

    issue(0);
    for (int t = 0; t < total; ++t) {
      if (t + 1 < total) {  // keep one DMA in flight while consuming chunk t
        issue(t + 1);
        __builtin_amdgcn_s_wait_tensorcnt(1);
      } else {
        __builtin_amdgcn_s_wait_tensorcnt(0);
      }
      __asm__ __volatile__("" ::: "memory");   // TDM wrote LDS behind the compiler's back
      const float* buf = (t & 1) ? buf1 : buf0;

      if (t < nch0) {                                   // ---- table 0, D = 64 ----
        int cnt = l0n - t * CHUNK; if (cnt > CHUNK) cnt = CHUNK;
        if (cnt == CHUNK) {
#pragma unroll
          for (int p = 0; p < CHUNK / 2; ++p) {
            const float4 v = *(const float4*)(buf + (2 * p + r0) * D0N + s0 * 4);
            a0x += v.x; a0y += v.y; a0z += v.z; a0w += v.w;
          }
        } else {
          const int passes = (cnt + 1) >> 1;
          for (int p = 0; p < passes; ++p) {
            const int r = 2 * p + r0;
            const float4 v = *(const float4*)(buf + r * D0N + s0 * 4);
            if (r < cnt) { a0x += v.x; a0y += v.y; a0z += v.z; a0w += v.w; }
          }
        }
      } else {                                          // ---- table 1, D = 128 ----
        int cnt = l1n - (t - nch0) * CHUNK; if (cnt > CHUNK) cnt = CHUNK;
        if (cnt == CHUNK) {
#pragma unroll
          for (int p = 0; p < CHUNK; ++p) {
            const float4 v = *(const float4*)(buf + p * D1N + lane * 4);
            a1x += v.x; a1y += v.y; a1z += v.z; a1w += v.w;
          }
        } else {
          for (int p = 0; p < cnt; ++p) {
            const float4 v = *(const float4*)(buf + p * D1N + lane * 4);
            a1x += v.x; a1y += v.y; a1z += v.z; a1w += v.w;
          }
        }
      }
      // WAR: next DMA reuses this buffer; TDM writes are unordered vs in-flight ds_loads.
      __asm__ __volatile__("s_wait_dscnt 0x0" ::: "memory");
    }
  }
#else
  for (int l = r0; l < l0n; l += 2) {
    const int row = idxr0[l];
    const float4 v = *(const float4*)(E0 + (size_t)row * D0N + s0 * 4);
    a0x += v.x; a0y += v.y; a0z += v.z; a0w += v.w;
  }
  for (int l = 0; l < l1n; ++l) {
    const int row = idxr1[l];
    const float4 v = *(const float4*)(E1 + (size_t)row * D1N + lane * 4);
    a1x += v.x; a1y += v.y; a1z += v.z; a1w += v.w;
  }
#endif

  float* outb = out + (size_t)b * OUTD;

  // table 0: fold the two half-wave partial sums (same 64 cols, alternating rows)
  a0x += __shfl_xor(a0x, 16, 32);
  a0y += __shfl_xor(a0y, 16, 32);
  a0z += __shfl_xor(a0z, 16, 32);
  a0w += __shfl_xor(a0w, 16, 32);
  if (lane < 16) {
    float4 o; o.x = a0x; o.y = a0y; o.z = a0z; o.w = a0w;
    *(float4*)(outb + s0 * 4) = o;
  }
  // table 1: 32 lanes x float4 = 128 cols
  {
    float4 o; o.x = a1x; o.y = a1y; o.z = a1z; o.w = a1w;
    *(float4*)(outb + D0N + lane * 4) = o;
  }
}

extern "C" void kernel_launch(void* const* d_in, const int* in_sizes, int n_in,
                              void* d_out, int out_size, void* d_ws, size_t ws_size,
                              hipStream_t stream) {
  (void)in_sizes; (void)n_in; (void)out_size; (void)d_ws; (void)ws_size;
  const int*   idx0 = (const int*)d_in[0];
  const int*   len0 = (const int*)d_in[1];
  const int*   idx1 = (const int*)d_in[2];
  const int*   len1 = (const int*)d_in[3];
  const float* E0   = (const float*)d_in[4];
  const float* E1   = (const float*)d_in[5];
  float* out = (float*)d_out;

  dim3 grid((B_TOTAL + WAVES_PER_BLOCK - 1) / WAVES_PER_BLOCK);  // 2048 blocks
  dim3 block(WAVES_PER_BLOCK * 32);                              // 256 threads = 8 waves
  embbag_tdm_kernel<<<grid, block, 0, stream>>>(idx0, len0, idx1, len1, E0, E1, out);
}